// GraphSageLayer_38010460569664
// MI455X (gfx1250) — compile-verified
//
#include <hip/hip_runtime.h>

typedef __attribute__((ext_vector_type(2))) float v2f;
typedef __attribute__((ext_vector_type(8))) float v8f;

#define N_NODES 100000
#define D 128
#define N_EDGES 640000
#define LDST 132   // LDS row stride in floats: 16B aligned rows, 132%64==4 -> conflict-free b64 lane pattern

// ---------------------------------------------------------------- kernel 1
__global__ void zero_ws_kernel(float4* __restrict__ p, int n4) {
    int i = blockIdx.x * blockDim.x + threadIdx.x;
    if (i < n4) p[i] = make_float4(0.f, 0.f, 0.f, 0.f);
}

// ---------------------------------------------------------------- kernel 2
// 32 threads per edge; each thread handles 4 consecutive features.
__global__ void __launch_bounds__(256)
scatter_edges_kernel(const float* __restrict__ x, const int* __restrict__ ei,
                     float* __restrict__ agg, float* __restrict__ cnt) {
    unsigned t = blockIdx.x * blockDim.x + threadIdx.x;
    unsigned e = t >> 5;
    unsigned c = t & 31;
    if (e >= N_EDGES) return;
    int src = ei[e];             // edge_index[0][e]
    int dst = ei[N_EDGES + e];   // edge_index[1][e]
    const float4 v = *(const float4*)(x + (size_t)src * D + c * 4);
    float* a = agg + (size_t)dst * D + c * 4;
    atomicAdd(a + 0, v.x);
    atomicAdd(a + 1, v.y);
    atomicAdd(a + 2, v.z);
    atomicAdd(a + 3, v.w);
    if (c == 0) atomicAdd(cnt + dst, 1.0f);
}

// ---------------------------------------------------------------- kernel 3
// One block of 128 threads (4 wave32) handles 16 output rows.
// Each wave computes two 16x16 tiles via V_WMMA_F32_16X16X4_F32 (fp32 path).
__global__ void __launch_bounds__(128)
sage_gemm_norm_kernel(const float* __restrict__ x, const float* __restrict__ agg,
                      const float* __restrict__ cnt, const float* __restrict__ Wl,
                      const float* __restrict__ Wr, const float* __restrict__ bias,
                      float* __restrict__ out) {
    __shared__ float sm[16 * LDST];   // mean rows
    __shared__ float sx[16 * LDST];   // x rows
    __shared__ float sinv[16];        // 1/max(cnt,1)
    __shared__ float srow[16];        // per-row sum of squares

    const int tid = threadIdx.x;
    const int rowbase = blockIdx.x * 16;

    if (tid < 16) {
        sinv[tid] = 1.0f / fmaxf(cnt[rowbase + tid], 1.0f);
        srow[tid] = 0.0f;
    }
    __syncthreads();

    // Stage mean (= agg * inv) and x into LDS. 512 float4 loads, 4 per thread.
    for (int idx = tid; idx < 512; idx += 128) {
        int r  = idx >> 5;   // row 0..15
        int c4 = idx & 31;   // float4 column 0..31
        size_t g = (size_t)(rowbase + r) * D + c4 * 4;
        float4 a  = *(const float4*)(agg + g);
        float4 xv = *(const float4*)(x + g);
        float inv = sinv[r];
        float* pm = &sm[r * LDST + c4 * 4];
        pm[0] = a.x * inv; pm[1] = a.y * inv; pm[2] = a.z * inv; pm[3] = a.w * inv;
        *(float4*)&sx[r * LDST + c4 * 4] = xv;
    }
    __syncthreads();

    const int lane = tid & 31;
    const int wave = tid >> 5;
    const int half = lane >> 4;   // 0: lanes 0-15, 1: lanes 16-31
    const int lm   = lane & 15;

    v8f acc[2];
    #pragma unroll
    for (int t = 0; t < 2; ++t) {
        const int colbase = (wave * 2 + t) * 16;
        v8f c = {};
        // mean @ W_l : K = 128 in steps of 4
        #pragma unroll 4
        for (int k = 0; k < D; k += 4) {
            // A 16x4 f32 layout: lanes 0-15 hold K=k,k+1 ; lanes 16-31 hold K=k+2,k+3
            v2f a = *(const v2f*)&sm[lm * LDST + k + 2 * half];
            v2f bf;
            bf[0] = Wl[(size_t)(k + 2 * half) * D + colbase + lm];
            bf[1] = Wl[(size_t)(k + 2 * half + 1) * D + colbase + lm];
            c = __builtin_amdgcn_wmma_f32_16x16x4_f32(false, a, false, bf,
                                                      (short)0, c, false, false);
        }
        // x @ W_r
        #pragma unroll 4
        for (int k = 0; k < D; k += 4) {
            v2f a = *(const v2f*)&sx[lm * LDST + k + 2 * half];
            v2f bf;
            bf[0] = Wr[(size_t)(k + 2 * half) * D + colbase + lm];
            bf[1] = Wr[(size_t)(k + 2 * half + 1) * D + colbase + lm];
            c = __builtin_amdgcn_wmma_f32_16x16x4_f32(false, a, false, bf,
                                                      (short)0, c, false, false);
        }
        // bias (same column for all 8 regs of this lane)
        const float bv = bias[colbase + lm];
        #pragma unroll
        for (int r = 0; r < 8; ++r) c[r] += bv;
        acc[t] = c;
        // per-row sum of squares: C/D layout -> reg r = row r (lanes 0-15),
        // row r+8 (lanes 16-31); reduce across 16 lanes of each half.
        #pragma unroll
        for (int r = 0; r < 8; ++r) {
            float v = c[r] * c[r];
            v += __shfl_xor(v, 1, 32);
            v += __shfl_xor(v, 2, 32);
            v += __shfl_xor(v, 4, 32);
            v += __shfl_xor(v, 8, 32);
            if (lm == 0) atomicAdd(&srow[r + 8 * half], v);
        }
    }
    __syncthreads();

    #pragma unroll
    for (int t = 0; t < 2; ++t) {
        const int colbase = (wave * 2 + t) * 16;
        #pragma unroll
        for (int r = 0; r < 8; ++r) {
            const int rl = r + 8 * half;
            const float scale = 1.0f / fmaxf(sqrtf(srow[rl]), 1e-12f);
            out[(size_t)(rowbase + rl) * D + colbase + lm] = acc[t][r] * scale;
        }
    }
}

// ---------------------------------------------------------------- launch
extern "C" void kernel_launch(void* const* d_in, const int* in_sizes, int n_in,
                              void* d_out, int out_size, void* d_ws, size_t ws_size,
                              hipStream_t stream) {
    const float* x  = (const float*)d_in[0];
    const int*   ei = (const int*)d_in[1];   // [2, N_EDGES] row-major
    const float* Wl = (const float*)d_in[2];
    const float* Wr = (const float*)d_in[3];
    const float* b  = (const float*)d_in[4];
    float* out = (float*)d_out;

    float* agg = (float*)d_ws;                       // N_NODES*D floats
    float* cnt = agg + (size_t)N_NODES * D;          // N_NODES floats

    const int n_zero  = N_NODES * D + N_NODES;       // 12,900,000 (divisible by 4)
    const int n4      = n_zero / 4;
    zero_ws_kernel<<<(n4 + 255) / 256, 256, 0, stream>>>((float4*)agg, n4);

    const int scatter_threads = N_EDGES * 32;        // 20,480,000
    scatter_edges_kernel<<<scatter_threads / 256, 256, 0, stream>>>(x, ei, agg, cnt);

    sage_gemm_norm_kernel<<<N_NODES / 16, 128, 0, stream>>>(x, agg, cnt, Wl, Wr, b, out);
}